// PreCrimeModel_76931454206189
// MI455X (gfx1250) — compile-verified
//
#include <hip/hip_runtime.h>

typedef __attribute__((ext_vector_type(2))) float v2f;
typedef __attribute__((ext_vector_type(8))) float v8f;

// ---------------------------------------------------------------------------
// Native f32 atomic add (global_atomic_add_f32, no-return variant when result
// is discarded).
// ---------------------------------------------------------------------------
__device__ __forceinline__ void fatomic_add(float* p, float v) {
    unsafeAtomicAdd(p, v);
}

// ---------------------------------------------------------------------------
// WMMA helper: D = A(16x4 f32) x B(4x16 f32) + C(16x16 f32)
// 8-arg pattern: (neg_a, A, neg_b, B, c_mod, C, reuse_a, reuse_b)
// ---------------------------------------------------------------------------
__device__ __forceinline__ v8f wmma4(v2f a, v2f b, v8f c) {
    return __builtin_amdgcn_wmma_f32_16x16x4_f32(
        false, a, false, b, (short)0, c, false, false);
}

// ---------------------------------------------------------------------------
// Zero workspace
// ---------------------------------------------------------------------------
__global__ void zero_kernel(float* __restrict__ p, long n) {
    long i = (long)blockIdx.x * blockDim.x + threadIdx.x;
    long stride = (long)gridDim.x * blockDim.x;
    long n4 = n >> 2;
    float4* p4 = reinterpret_cast<float4*>(p);
    for (long j = i; j < n4; j += stride)
        p4[j] = make_float4(0.f, 0.f, 0.f, 0.f);
    for (long j = (n4 << 2) + i; j < n; j += stride)
        p[j] = 0.f;
}

// ---------------------------------------------------------------------------
// Layer-1 edge scatter: sum[dst] += x_src[src] (F=4) and cnt[dst] += 1
// ---------------------------------------------------------------------------
__global__ __launch_bounds__(256) void scatter_l1(
    const float* __restrict__ x_src,
    const int* __restrict__ src, const int* __restrict__ dst,
    float* __restrict__ sum, float* __restrict__ cnt, int E) {
    int e = blockIdx.x * blockDim.x + threadIdx.x;
    if (e >= E) return;
    int s = src[e], d = dst[e];
    float4 v = reinterpret_cast<const float4*>(x_src)[s];
    float* b = sum + (size_t)d * 4;
    fatomic_add(b + 0, v.x);
    fatomic_add(b + 1, v.y);
    fatomic_add(b + 2, v.z);
    fatomic_add(b + 3, v.w);
    fatomic_add(cnt + d, 1.0f);
}

// ---------------------------------------------------------------------------
// Layer-2 edge scatter: sum[dst] += h_src[src] (F=16)
// ---------------------------------------------------------------------------
__global__ __launch_bounds__(256) void scatter_l2(
    const float* __restrict__ h_src,
    const int* __restrict__ src, const int* __restrict__ dst,
    float* __restrict__ sum, int E) {
    int e = blockIdx.x * blockDim.x + threadIdx.x;
    if (e >= E) return;
    int s = src[e], d = dst[e];
    const float4* hv = reinterpret_cast<const float4*>(h_src + (size_t)s * 16);
    float* b = sum + (size_t)d * 16;
#pragma unroll
    for (int q = 0; q < 4; ++q) {
        float4 v = hv[q];
        fatomic_add(b + 4 * q + 0, v.x);
        fatomic_add(b + 4 * q + 1, v.y);
        fatomic_add(b + 4 * q + 2, v.z);
        fatomic_add(b + 4 * q + 3, v.w);
    }
}

// ---------------------------------------------------------------------------
// Layer-1 node transform: h = relu((sum/max(cnt,1)) @ Wl + bl + x_dst @ Wr)
// One wave (32 lanes) computes a 16-node x 16-feature tile with two chained
// V_WMMA_F32_16X16X4_F32 ops.
// A layout (16x4 f32): lane<16 -> M=lane, K={0,1}; lane>=16 -> M=lane-16, K={2,3}
// B layout (4x16 f32): mirrors A: VGPR0 = K{0|2}, VGPR1 = K{1|3}, N = lane&15
// C layout: VGPR r -> M = r + (lane>>4)*8, N = lane&15
// ---------------------------------------------------------------------------
__global__ __launch_bounds__(256) void l1_transform(
    const float* __restrict__ sum1, const float* __restrict__ cnt,
    const float* __restrict__ x_dst,
    const float* __restrict__ Wl, const float* __restrict__ bl,
    const float* __restrict__ Wr,
    float* __restrict__ h_out, int n_tiles) {
    int wave = threadIdx.x >> 5;
    int tile = blockIdx.x * (blockDim.x >> 5) + wave;
    if (tile >= n_tiles) return;           // wave-uniform: EXEC all-1s for WMMA
    int lane = threadIdx.x & 31;
    int m  = lane & 15;                    // M for A-loads, N for B/C
    int hi = lane >> 4;
    int kb = hi * 2;
    int node = tile * 16 + m;

    float c = fmaxf(cnt[node], 1.0f);
    v2f a_mean, a_root, b_l, b_r;
    a_mean.x = sum1[node * 4 + kb] / c;
    a_mean.y = sum1[node * 4 + kb + 1] / c;
    a_root.x = x_dst[node * 4 + kb];
    a_root.y = x_dst[node * 4 + kb + 1];
    b_l.x = Wl[kb * 16 + m];
    b_l.y = Wl[(kb + 1) * 16 + m];
    b_r.x = Wr[kb * 16 + m];
    b_r.y = Wr[(kb + 1) * 16 + m];

    float bias = bl[m];
    v8f acc;
#pragma unroll
    for (int r = 0; r < 8; ++r) acc[r] = bias;

    acc = wmma4(a_mean, b_l, acc);
    acc = wmma4(a_root, b_r, acc);

    float* out = h_out + ((size_t)tile * 16 + hi * 8) * 16 + m;
#pragma unroll
    for (int r = 0; r < 8; ++r)
        out[r * 16] = fmaxf(acc[r], 0.0f);
}

// ---------------------------------------------------------------------------
// Layer-2 node transform: z = relu((sum/max(cnt,1)) @ Wl + bl + h_root @ Wr)
// K=16 via 8 chained K=4 WMMAs. May run in place (z_out == h_root): each wave
// only touches its own 16-node tile and all loads precede stores.
// ---------------------------------------------------------------------------
__global__ __launch_bounds__(256) void l2_transform(
    const float* __restrict__ sum2, const float* __restrict__ cnt,
    const float* __restrict__ h_root,
    const float* __restrict__ Wl, const float* __restrict__ bl,
    const float* __restrict__ Wr,
    float* __restrict__ z_out, int n_tiles) {
    int wave = threadIdx.x >> 5;
    int tile = blockIdx.x * (blockDim.x >> 5) + wave;
    if (tile >= n_tiles) return;
    int lane = threadIdx.x & 31;
    int m  = lane & 15;
    int hi = lane >> 4;
    int node = tile * 16 + m;

    float c = fmaxf(cnt[node], 1.0f);
    const float* srow = sum2 + (size_t)node * 16;
    const float* rrow = h_root + (size_t)node * 16;

    float bias = bl[m];
    v8f acc;
#pragma unroll
    for (int r = 0; r < 8; ++r) acc[r] = bias;

#pragma unroll
    for (int s = 0; s < 4; ++s) {
        int kb = 4 * s + 2 * hi;
        v2f a, b;
        a.x = srow[kb] / c;
        a.y = srow[kb + 1] / c;
        b.x = Wl[kb * 16 + m];
        b.y = Wl[(kb + 1) * 16 + m];
        acc = wmma4(a, b, acc);
    }
#pragma unroll
    for (int s = 0; s < 4; ++s) {
        int kb = 4 * s + 2 * hi;
        v2f a, b;
        a.x = rrow[kb];
        a.y = rrow[kb + 1];
        b.x = Wr[kb * 16 + m];
        b.y = Wr[(kb + 1) * 16 + m];
        acc = wmma4(a, b, acc);
    }

    float* out = z_out + ((size_t)tile * 16 + hi * 8) * 16 + m;
#pragma unroll
    for (int r = 0; r < 8; ++r)
        out[r * 16] = fmaxf(acc[r], 0.0f);
}

// ---------------------------------------------------------------------------
// Host-side launch
// ---------------------------------------------------------------------------
extern "C" void kernel_launch(void* const* d_in, const int* in_sizes, int n_in,
                              void* d_out, int out_size, void* d_ws, size_t ws_size,
                              hipStream_t stream) {
    const float* x_p    = (const float*)d_in[0];
    const float* x_u    = (const float*)d_in[1];
    const int*   src_pu = (const int*)d_in[2];
    const int*   dst_pu = (const int*)d_in[3];
    const int*   src_up = (const int*)d_in[4];
    const int*   dst_up = (const int*)d_in[5];
    const float* Wl1_pu = (const float*)d_in[6];
    const float* bl1_pu = (const float*)d_in[7];
    const float* Wr1_pu = (const float*)d_in[8];
    const float* Wl1_up = (const float*)d_in[9];
    const float* bl1_up = (const float*)d_in[10];
    const float* Wr1_up = (const float*)d_in[11];
    const float* Wl2_pu = (const float*)d_in[12];
    const float* bl2_pu = (const float*)d_in[13];
    const float* Wr2_pu = (const float*)d_in[14];
    const float* Wl2_up = (const float*)d_in[15];
    const float* bl2_up = (const float*)d_in[16];
    const float* Wr2_up = (const float*)d_in[17];

    const int NP  = in_sizes[0] / 4;
    const int NU  = in_sizes[1] / 4;
    const int E1  = in_sizes[2];   // pu edges
    const int E2  = in_sizes[4];   // up edges

    // Workspace layout (floats)
    float* ws     = (float*)d_ws;
    float* cnt_u  = ws;                              // NU
    float* cnt_p  = cnt_u + NU;                      // NP
    float* sum1_u = cnt_p + NP;                      // NU*4
    float* sum1_p = sum1_u + (size_t)NU * 4;         // NP*4
    float* sum2_u = sum1_p + (size_t)NP * 4;         // NU*16
    float* sum2_p = sum2_u + (size_t)NU * 16;        // NP*16
    long ws_floats = (long)(sum2_p + (size_t)NP * 16 - ws);

    // d_out: z_p [NP*16] then z_u [NU*16]; also stages h_p / h_u.
    float* z_p = (float*)d_out;
    float* z_u = z_p + (size_t)NP * 16;

    // 1) zero accumulators
    zero_kernel<<<2048, 256, 0, stream>>>(ws, ws_floats);

    // 2) layer-1 edge scatter (+ degree counts, reused by layer 2)
    int eb1 = (E1 + 255) / 256;
    int eb2 = (E2 + 255) / 256;
    scatter_l1<<<eb1, 256, 0, stream>>>(x_p, src_pu, dst_pu, sum1_u, cnt_u, E1);
    scatter_l1<<<eb2, 256, 0, stream>>>(x_u, src_up, dst_up, sum1_p, cnt_p, E2);

    // 3) layer-1 node transforms (WMMA); h_u -> z_u region, h_p -> z_p region
    int tu = NU / 16, tp = NP / 16;
    l1_transform<<<(tu + 7) / 8, 256, 0, stream>>>(sum1_u, cnt_u, x_u,
                                                   Wl1_pu, bl1_pu, Wr1_pu, z_u, tu);
    l1_transform<<<(tp + 7) / 8, 256, 0, stream>>>(sum1_p, cnt_p, x_p,
                                                   Wl1_up, bl1_up, Wr1_up, z_p, tp);

    // 4) layer-2 edge scatter (reads staged h_p / h_u)
    scatter_l2<<<eb1, 256, 0, stream>>>(z_p /*h_p*/, src_pu, dst_pu, sum2_u, E1);
    scatter_l2<<<eb2, 256, 0, stream>>>(z_u /*h_u*/, src_up, dst_up, sum2_p, E2);

    // 5) layer-2 node transforms (WMMA, in place over staged h)
    l2_transform<<<(tu + 7) / 8, 256, 0, stream>>>(sum2_u, cnt_u, z_u,
                                                   Wl2_pu, bl2_pu, Wr2_pu, z_u, tu);
    l2_transform<<<(tp + 7) / 8, 256, 0, stream>>>(sum2_p, cnt_p, z_p,
                                                   Wl2_up, bl2_up, Wr2_up, z_p, tp);
}